// WassersteinLoss_4346506903751
// MI455X (gfx1250) — compile-verified
//
#include <hip/hip_runtime.h>
#include <hip/hip_bf16.h>

// ---------------------------------------------------------------------------
// WassersteinLoss: out = mean_b[ 2/(T*C*(T+1)) * sum_{c,t} |yp-yt| * (T-t) ]
// B=32, C=64, T=4096  -> pure bandwidth-bound weighted-abs reduction (67 MB).
// Pass 1: streaming float4 grid-stride partial sums (one per block -> d_ws).
// Pass 2: single block reduces partials; last 32->1 step uses
//         V_WMMA_F32_16X16X4_F32 with an all-ones B matrix.
// ---------------------------------------------------------------------------

#define NBLOCKS  2048
#define NTHREADS 256

typedef __attribute__((ext_vector_type(2))) float v2f;
typedef __attribute__((ext_vector_type(8))) float v8f;

constexpr int kB = 32;
constexpr int kC = 64;
constexpr int kT = 4096;

__global__ __launch_bounds__(NTHREADS)
void wass_partial_kernel(const float* __restrict__ yp,
                         const float* __restrict__ yt,
                         float* __restrict__ partials,
                         int n)
{
    const int tid    = blockIdx.x * NTHREADS + threadIdx.x;
    const int stride = gridDim.x * NTHREADS;
    const int n4     = n >> 2;

    const float4* __restrict__ p4 = reinterpret_cast<const float4*>(yp);
    const float4* __restrict__ t4 = reinterpret_cast<const float4*>(yt);

    float acc = 0.0f;
    for (int i = tid; i < n4; i += stride) {
        float4 a = p4[i];               // global_load_b128
        float4 b = t4[i];               // global_load_b128
        // flat element index = 4*i ; t = index mod T (T=4096, vec4-aligned rows)
        int   t0 = (i << 2) & (kT - 1);
        float w0 = (float)(kT - t0);    // weights w(t) = T - t
        acc = fmaf(fabsf(a.x - b.x), w0,        acc);
        acc = fmaf(fabsf(a.y - b.y), w0 - 1.0f, acc);
        acc = fmaf(fabsf(a.z - b.z), w0 - 2.0f, acc);
        acc = fmaf(fabsf(a.w - b.w), w0 - 3.0f, acc);
    }

    // wave32 reduction
    for (int off = 16; off > 0; off >>= 1)
        acc += __shfl_down(acc, off, 32);

    __shared__ float smem[NTHREADS / 32];   // 8 waves per block
    const int lane = threadIdx.x & 31;
    const int wid  = threadIdx.x >> 5;
    if (lane == 0) smem[wid] = acc;
    __syncthreads();

    if (wid == 0) {
        float v = (lane < (NTHREADS / 32)) ? smem[lane] : 0.0f;
        for (int off = 4; off > 0; off >>= 1)
            v += __shfl_down(v, off, 32);
        if (lane == 0) partials[blockIdx.x] = v;
    }
}

__global__ __launch_bounds__(256)
void wass_final_kernel(const float* __restrict__ partials,
                       float* __restrict__ out,
                       int np, float scale)
{
    __shared__ float smem[256];
    float acc = 0.0f;
    for (int i = threadIdx.x; i < np; i += 256)
        acc += partials[i];
    smem[threadIdx.x] = acc;
    __syncthreads();

    const int lane = threadIdx.x & 31;
    const int wid  = threadIdx.x >> 5;

    if (wid == 0) {                      // wave-uniform branch: EXEC all-ones
        // 256 block partials -> 32 lane values
        float v = 0.0f;
        #pragma unroll
        for (int j = 0; j < 8; ++j)
            v += smem[lane + 32 * j];

#if __has_builtin(__builtin_amdgcn_wmma_f32_16x16x4_f32)
        // A (16x4 f32): VGPR0 = lane partials, VGPR1 = 0.
        //   A[m,0]=v[lane m], A[m,2]=v[lane 16+m], A[m,1]=A[m,3]=0.
        // B (4x16) = all ones  ->  D[m,n] = sum_k A[m,k]  (layout-independent).
        v2f a; a.x = v;    a.y = 0.0f;
        v2f b; b.x = 1.0f; b.y = 1.0f;
        v8f c = {};
        c = __builtin_amdgcn_wmma_f32_16x16x4_f32(
                /*neg_a=*/false, a, /*neg_b=*/false, b,
                /*c_mod=*/(short)0, c, /*reuse_a=*/false, /*reuse_b=*/false);
        // lane L<16 holds D[0..7, L]; lane L>=16 holds D[8..15, L-16].
        float lsum = c[0] + c[1] + c[2] + c[3] + c[4] + c[5] + c[6] + c[7];
        lsum += __shfl_xor(lsum, 16, 32);   // rows 0..7 + rows 8..15
#else
        // scalar fallback (keeps compile green if builtin signature differs)
        float lsum = v;
        for (int off = 16; off > 0; off >>= 1)
            lsum += __shfl_down(lsum, off, 32);
        lsum = __shfl(lsum, 0, 32);
#endif
        if (lane == 0) out[0] = lsum * scale;
    }
}

extern "C" void kernel_launch(void* const* d_in, const int* in_sizes, int n_in,
                              void* d_out, int out_size, void* d_ws, size_t ws_size,
                              hipStream_t stream)
{
    const float* yp = (const float*)d_in[0];
    const float* yt = (const float*)d_in[1];
    float* out      = (float*)d_out;
    float* partials = (float*)d_ws;          // NBLOCKS floats = 8 KB scratch
    const int n     = in_sizes[0];           // B*C*T = 8,388,608

    wass_partial_kernel<<<NBLOCKS, NTHREADS, 0, stream>>>(yp, yt, partials, n);

    // scale = (2 / (T*C*(T+1))) / B, computed in double on host for accuracy
    const double scaling = 2.0 / ((double)kT * (double)kC * ((double)kT + 1.0));
    const float  scale   = (float)(scaling / (double)kB);

    wass_final_kernel<<<1, 256, 0, stream>>>(partials, out, NBLOCKS, scale);
}